// MolAttention_29489245455068
// MI455X (gfx1250) — compile-verified
//
#include <hip/hip_runtime.h>
#include <math.h>

// ---------------------------------------------------------------------------
// MolAttention GAT pipeline for gfx1250 (MI455X).
// Dense GEMMs: V_WMMA_F32_16X16X4_F32 with LDS-staged, fragment-permuted
// A/B panels (conflict-free ds_load_b64 per fragment, 8x B-fragment reuse).
// The edge phase is HBM/atomic bound (~2 GB traffic); GEMMs (~15 GFLOP fp32)
// are nowhere near the WMMA ceiling, so fp32-exact is the right precision.
// ---------------------------------------------------------------------------

#define HID 128
#define NEG_SLOPE 0.01f
#define TB 256

typedef __attribute__((ext_vector_type(2))) float v2f;
typedef __attribute__((ext_vector_type(8))) float v8f;

// Monotonic order-preserving float->uint encoding (so atomicMax(u32) == float max).
// All finite floats encode to >= 0x00800000, so enc==0 means "never touched".
__device__ __forceinline__ unsigned fenc(float x) {
  unsigned b = __float_as_uint(x);
  return (b & 0x80000000u) ? ~b : (b | 0x80000000u);
}
__device__ __forceinline__ float fdec_or0(unsigned u) {
  if (u == 0u) return 0.0f;                      // segment had no edges -> max := 0
  unsigned b = (u & 0x80000000u) ? (u & 0x7FFFFFFFu) : ~u;
  return __uint_as_float(b);
}

// ---------------------------------------------------------------------------
// GEMM: C[M x 128] = A[M x K] @ B[K x 128], fp32 WMMA 16x16x4.
// Block = 8 waves; 128-row M-panel per block; wave w owns 16-col N-tile w.
// A-panel (128 x K) and full B (K x 128) are staged in dynamic LDS in WMMA
// fragment order:
//   A frag (mtile,kstep): lane L holds A[m0+16*mtile+L%16][4*kstep+2*(L/16)+{0,1}]
//   B frag (ntile,kstep): lane L holds B[4*kstep+2*(L/16)+{0,1}][16*ntile+L%16]
// so each fragment fetch is one conflict-free ds_load_b64.
// Per k-step: 1 B-frag load reused over 8 M-subtiles (8 A-frag loads + 8 WMMA).
// act==1: fused bias + leaky_relu epilogue.
// ---------------------------------------------------------------------------
__global__ __launch_bounds__(TB)
void gemm128_wmma_f32(const float* __restrict__ A, const float* __restrict__ B,
                      const float* __restrict__ bias, float* __restrict__ C,
                      int M, int K, int act) {
  extern __shared__ float lds[];
  const int ksteps = K >> 2;                         // K/4 WMMA steps
  float* sA = lds;                                   // 8*ksteps*32*2 floats
  float* sB = lds + (size_t)8 * ksteps * 64;         // 8*ksteps*32*2 floats
  const int tid  = threadIdx.x;
  const int lane = tid & 31;
  const int wave = tid >> 5;                         // 0..7 -> N-tile
  const int m0   = blockIdx.x * 128;

  // ---- stage A panel (rows m0..m0+127), coalesced float4 reads, permuted ----
  for (int e = tid * 4; e < 128 * K; e += TB * 4) {
    const int m = e / K;
    const int k = e - m * K;                         // multiple of 4
    int row = m0 + m;
    if (row >= M) row = M - 1;                       // clamp: junk rows never stored
    const float4 v = *(const float4*)(A + (size_t)row * K + k);
    const int mtile = m >> 4, mr = m & 15, kstep = k >> 2;
    float* base = sA + (size_t)((mtile * ksteps + kstep) * 32) * 2;
    base[(mr)      * 2 + 0] = v.x;                   // K%4==0 -> lanes 0-15
    base[(mr)      * 2 + 1] = v.y;                   // K%4==1
    base[(mr + 16) * 2 + 0] = v.z;                   // K%4==2 -> lanes 16-31
    base[(mr + 16) * 2 + 1] = v.w;                   // K%4==3
  }
  // ---- stage full B (K x 128), coalesced float4 reads, permuted ----
  for (int e = tid * 4; e < K * 128; e += TB * 4) {
    const int kr = e >> 7, c = e & 127;              // c multiple of 4
    const float4 v = *(const float4*)(B + (size_t)kr * HID + c);
    const int kstep = kr >> 2, r = kr & 3;
    const int slot = r & 1, l0 = (c & 15) + 16 * (r >> 1);
    float* base = sB + (size_t)(((c >> 4) * ksteps + kstep) * 32) * 2;
    base[(l0 + 0) * 2 + slot] = v.x;
    base[(l0 + 1) * 2 + slot] = v.y;
    base[(l0 + 2) * 2 + slot] = v.z;
    base[(l0 + 3) * 2 + slot] = v.w;
  }
  __syncthreads();

  // ---- compute 8 x (16x16) accumulators from LDS fragments ----
  v8f acc[8];
#pragma unroll
  for (int mt = 0; mt < 8; ++mt) acc[mt] = (v8f){};

  const v2f* aF = (const v2f*)sA;
  const v2f* bF = (const v2f*)sB + (size_t)wave * ksteps * 32;
  for (int ks = 0; ks < ksteps; ++ks) {
    const v2f b = bF[ks * 32 + lane];                // 1 ds_load_b64, reused 8x
#pragma unroll
    for (int mt = 0; mt < 8; ++mt) {
      const v2f a = aF[(mt * ksteps + ks) * 32 + lane];
      acc[mt] = __builtin_amdgcn_wmma_f32_16x16x4_f32(
          false, a, false, b, (short)0, acc[mt], false, false);
    }
  }

  // ---- epilogue ----
  const int half = lane >> 4, l15 = lane & 15;
  const int col  = wave * 16 + l15;
  const float bv = (act == 1) ? bias[col] : 0.0f;
#pragma unroll
  for (int mt = 0; mt < 8; ++mt) {
#pragma unroll
    for (int i = 0; i < 8; ++i) {
      const int r = m0 + mt * 16 + half * 8 + i;
      if (r < M) {
        float v = acc[mt][i];
        if (act == 1) { v += bv; v = v > 0.0f ? v : NEG_SLOPE * v; }
        C[(size_t)r * HID + col] = v;
      }
    }
  }
}

// Per-row dual dot products: outS[n] = h[n,:].aS, outD[n] = h[n,:].aD. Wave/row.
__global__ __launch_bounds__(TB)
void node_dots_k(const float* __restrict__ h, const float* __restrict__ aS,
                 const float* __restrict__ aD, float* __restrict__ outS,
                 float* __restrict__ outD, int n) {
  const int wid  = (int)((blockIdx.x * 256u + threadIdx.x) >> 5);
  const int lane = threadIdx.x & 31;
  if (wid >= n) return;
  const float* row = h + (size_t)wid * HID;
  float s0 = 0.0f, s1 = 0.0f;
#pragma unroll
  for (int j = 0; j < HID; j += 32) {
    const float v = row[j + lane];
    s0 += v * aS[j + lane];
    s1 += v * aD[j + lane];
  }
  for (int o = 16; o; o >>= 1) { s0 += __shfl_xor(s0, o, 32); s1 += __shfl_xor(s1, o, 32); }
  if (lane == 0) { outS[wid] = s0; outD[wid] = s1; }
}

__global__ void zero_b32_k(unsigned* p, long n) {
  const long i = (long)blockIdx.x * TB + threadIdx.x;
  if (i < n) p[i] = 0u;
}

// e = leaky(as[src] + ad[dst]); segment max via encoded atomicMax.
__global__ void edge_max_k(const int* __restrict__ src, const int* __restrict__ dst,
                           const float* __restrict__ as_, const float* __restrict__ ad_,
                           float* __restrict__ e, unsigned* __restrict__ smax,
                           int n, int ident) {
  const int i = blockIdx.x * TB + threadIdx.x;
  if (i >= n) return;
  const int s = ident ? i : src[i];
  const int d = dst[i];
  float v = as_[s] + ad_[d];
  v = v > 0.0f ? v : NEG_SLOPE * v;
  e[i] = v;
  atomicMax(&smax[d], fenc(v));
}

// ex = exp(e - segmax[dst]); denom[dst] += ex (e buffer overwritten with ex).
__global__ void edge_exp_k(const int* __restrict__ dst, float* __restrict__ e,
                           const unsigned* __restrict__ smax, float* __restrict__ denom,
                           int n) {
  const int i = blockIdx.x * TB + threadIdx.x;
  if (i >= n) return;
  const int d = dst[i];
  const float ex = expf(e[i] - fdec_or0(smax[d]));
  e[i] = ex;
  atomicAdd(&denom[d], ex);
}

// acc[dst,:] += (ex/denom[dst]) * hs[src,:].  One wave per edge.
__global__ __launch_bounds__(TB)
void edge_scatter_k(const int* __restrict__ src, const int* __restrict__ dst,
                    const float* __restrict__ ex, const float* __restrict__ denom,
                    const float* __restrict__ hs, float* __restrict__ acc,
                    int n, int ident) {
  const int wid  = (int)((blockIdx.x * 256u + threadIdx.x) >> 5);
  const int lane = threadIdx.x & 31;
  if (wid >= n) return;
  const int s = ident ? wid : src[wid];
  const int d = dst[wid];
  const float alpha = ex[wid] / (denom[d] + 1e-16f);
  const float* hr = hs + (size_t)s * HID;
  float* ar = acc + (size_t)d * HID;
#pragma unroll
  for (int j = 0; j < HID; j += 32) atomicAdd(&ar[j + lane], alpha * hr[j + lane]);
}

// In-place h = elu(h + b[col]).
__global__ void bias_elu_k(float* __restrict__ h, const float* __restrict__ b, long n) {
  const long i = (long)blockIdx.x * TB + threadIdx.x;
  if (i >= n) return;
  const float v = h[i] + b[i & (HID - 1)];
  h[i] = v > 0.0f ? v : (expf(v) - 1.0f);
}

__global__ void relu_ip_k(float* __restrict__ h, long n) {
  const long i = (long)blockIdx.x * TB + threadIdx.x;
  if (i >= n) return;
  const float v = h[i];
  h[i] = v > 0.0f ? v : 0.0f;
}

// pooled[batch[n], :] += h[n, :]   (thread per element)
__global__ void pool_sum_k(const float* __restrict__ h, const int* __restrict__ batch,
                           float* __restrict__ pooled, long n) {
  const long i = (long)blockIdx.x * TB + threadIdx.x;
  if (i >= n) return;
  const int node = (int)(i >> 7);
  atomicAdd(&pooled[(size_t)batch[node] * HID + (i & (HID - 1))], h[i]);
}

// out[g] = dot(Gm[g,:], W2[:,0]) + b2[0].  Wave per graph.
__global__ __launch_bounds__(TB)
void final_gemv_k(const float* __restrict__ Gm, const float* __restrict__ W2,
                  const float* __restrict__ b2, float* __restrict__ out, int ng) {
  const int wid  = (int)((blockIdx.x * 256u + threadIdx.x) >> 5);
  const int lane = threadIdx.x & 31;
  if (wid >= ng) return;
  const float* r = Gm + (size_t)wid * HID;
  float s = 0.0f;
#pragma unroll
  for (int j = 0; j < HID; j += 32) s += r[j + lane] * W2[j + lane];
  for (int o = 16; o; o >>= 1) s += __shfl_xor(s, o, 32);
  if (lane == 0) out[wid] = s + b2[0];
}

// ---------------------------------------------------------------------------
extern "C" void kernel_launch(void* const* d_in, const int* in_sizes, int n_in,
                              void* d_out, int out_size, void* d_ws, size_t ws_size,
                              hipStream_t stream) {
  const float* x      = (const float*)d_in[0];
  const int*   eidx   = (const int*)d_in[1];
  const int*   batch  = (const int*)d_in[2];
  const float* W1     = (const float*)d_in[3];
  const float* b1     = (const float*)d_in[4];
  const float* gW     = (const float*)d_in[5];
  const float* g_asrc = (const float*)d_in[6];
  const float* g_adst = (const float*)d_in[7];
  const float* g_b    = (const float*)d_in[8];
  const float* mW     = (const float*)d_in[9];
  const float* m_asrc = (const float*)d_in[10];
  const float* m_adst = (const float*)d_in[11];
  const float* m_b    = (const float*)d_in[12];
  const float* W2     = (const float*)d_in[13];
  const float* b2     = (const float*)d_in[14];
  float* out = (float*)d_out;

  const int FIN = 64;
  const int N = in_sizes[0] / FIN;      // 100000 nodes
  const int E = in_sizes[1] / 2;        // 400000 edges
  const int G = out_size;               // 5000 graphs (OUT=1)
  const int* src = eidx;
  const int* dst = eidx + E;

  // Workspace layout (~113 MB): two ping-pong N x H buffers + small arrays.
  float* p = (float*)d_ws;
  float* hA    = p; p += (size_t)N * HID;   // h (layer input / aggregation target)
  float* hB    = p; p += (size_t)N * HID;   // hs (W-transformed features)
  float* ebuf  = p; p += E;                 // per-edge score -> exp
  float* asn   = p; p += N;
  float* adn   = p; p += N;
  unsigned* smax = (unsigned*)p; p += N;    // encoded segment max
  float* denom = p; p += N;                 // segment sum (adjacent to smax)
  float* pooled= p; p += (size_t)G * HID;
  float* hd    = p; p += (size_t)G * HID;   // transformed pooled features
  float* accg  = p; p += (size_t)G * HID;   // graph-level aggregation
  float* asg   = p; p += G;
  float* adg   = p; p += G;
  (void)asg; (void)ws_size; (void)n_in;

  const long NH = (long)N * HID;
  const long GH = (long)G * HID;
  auto cdiv = [](long a, long b) { return (int)((a + b - 1) / b); };
  // dynamic LDS bytes for gemm: (sA + sB) = 2 * 8*ksteps*64 floats = 4096*(K/4) B
  const size_t lds64  = 4096u * (64 / 4);   //  64 KB (K=64)
  const size_t lds128 = 4096u * (128 / 4);  // 128 KB (K=128), 320 KB WGP has room

  // 1) h = leaky_relu(x @ W1 + b1)
  gemm128_wmma_f32<<<cdiv(N, 128), TB, lds64, stream>>>(x, W1, b1, hA, N, FIN, 1);

  // 2) three GAT layers (h in hA, hs in hB, aggregate back into hA)
  for (int l = 0; l < 3; ++l) {
    gemm128_wmma_f32<<<cdiv(N, 128), TB, lds128, stream>>>(
        hA, gW + (size_t)l * HID * HID, nullptr, hB, N, HID, 0);
    node_dots_k<<<cdiv(N, 8), TB, 0, stream>>>(hB, g_asrc + l * HID, g_adst + l * HID,
                                               asn, adn, N);
    zero_b32_k<<<cdiv(2L * N, TB), TB, 0, stream>>>(smax, 2L * N);  // smax + denom
    edge_max_k<<<cdiv(E, TB), TB, 0, stream>>>(src, dst, asn, adn, ebuf, smax, E, 0);
    edge_exp_k<<<cdiv(E, TB), TB, 0, stream>>>(dst, ebuf, smax, denom, E);
    zero_b32_k<<<cdiv(NH, TB), TB, 0, stream>>>((unsigned*)hA, NH);
    edge_scatter_k<<<cdiv(E, 8), TB, 0, stream>>>(src, dst, ebuf, denom, hB, hA, E, 0);
    bias_elu_k<<<cdiv(NH, TB), TB, 0, stream>>>(hA, g_b + l * HID, NH);
  }

  // 3) pooled = relu(segment_sum(h, batch))
  zero_b32_k<<<cdiv(GH, TB), TB, 0, stream>>>((unsigned*)pooled, GH);
  pool_sum_k<<<cdiv(NH, TB), TB, 0, stream>>>(hA, batch, pooled, NH);
  relu_ip_k<<<cdiv(GH, TB), TB, 0, stream>>>(pooled, GH);

  // 4) readout bipartite GAT: nodes -> graphs (src identity, dst = batch)
  gemm128_wmma_f32<<<cdiv(N, 128), TB, lds128, stream>>>(hA, mW, nullptr, hB, N, HID, 0);
  gemm128_wmma_f32<<<cdiv(G, 128), TB, lds128, stream>>>(pooled, mW, nullptr, hd, G, HID, 0);
  node_dots_k<<<cdiv(N, 8), TB, 0, stream>>>(hB, m_asrc, m_adst, asn, adn, N);
  node_dots_k<<<cdiv(G, 8), TB, 0, stream>>>(hd, m_asrc, m_adst, asg, adg, G);
  zero_b32_k<<<cdiv(2L * N, TB), TB, 0, stream>>>(smax, 2L * N);    // covers G too
  edge_max_k<<<cdiv(N, TB), TB, 0, stream>>>(nullptr, batch, asn, adg, ebuf, smax, N, 1);
  edge_exp_k<<<cdiv(N, TB), TB, 0, stream>>>(batch, ebuf, smax, denom, N);
  zero_b32_k<<<cdiv(GH, TB), TB, 0, stream>>>((unsigned*)accg, GH);
  edge_scatter_k<<<cdiv(N, 8), TB, 0, stream>>>(nullptr, batch, ebuf, denom, hB, accg, N, 1);
  bias_elu_k<<<cdiv(GH, TB), TB, 0, stream>>>(accg, m_b, GH);

  // 5) out = accg @ W2 + b2
  final_gemv_k<<<cdiv(G, 8), TB, 0, stream>>>(accg, W2, b2, out, G);
}